// LegalGraphNeuralNetwork_867583393904
// MI455X (gfx1250) — compile-verified
//
#include <hip/hip_runtime.h>
#include <math.h>

typedef __attribute__((ext_vector_type(2))) float v2f;
typedef __attribute__((ext_vector_type(8))) float v8f;

#define N_NODES 4096
#define DIM     768
#define E_EDGES 131072
#define NET     5
#define NL      3
#define TD      2304   // 3*DIM
#define NH      8
#define DH      96
#define CH      512    // attention key chunk held in LDS

// D = A(16x4) * B(4x16) + C, fp32 WMMA.
// Lane L: m = L&15, kh = L>>4. a0=A[m][k0+2kh], a1=A[m][k0+2kh+1]; b likewise for col n=m.
// D element j: row = j + 8*kh, col = L&15.
__device__ __forceinline__ v8f wmma4(float a0, float a1, float b0, float b1, v8f c) {
    v2f a; a.x = a0; a.y = a1;
    v2f b; b.x = b0; b.y = b1;
    return __builtin_amdgcn_wmma_f32_16x16x4_f32(false, a, false, b, (short)0, c, false, false);
}

// ---------------- embedding + add ----------------
__global__ void k_embed(const float* __restrict__ nf, const int* __restrict__ ntype,
                        const float* __restrict__ st, const float* __restrict__ ca,
                        const float* __restrict__ ar, float* __restrict__ x) {
    int row = blockIdx.x;
    int t = ntype[row];
    const float* tab; int mrow;
    if (t == 0)      { tab = st; mrow = row % 1000; }
    else if (t == 1) { tab = ca; mrow = row % 5000; }
    else             { tab = ar; mrow = row % 2000; }
    for (int c = threadIdx.x; c < DIM; c += blockDim.x)
        x[(size_t)row * DIM + c] = nf[(size_t)row * DIM + c] + tab[(size_t)mrow * DIM + c];
}

// ---------------- edge bucket scatter (one wave per edge) ----------------
__global__ void k_scatter(const float* __restrict__ x, const int* __restrict__ src,
                          const int* __restrict__ tgt, const int* __restrict__ et,
                          float* __restrict__ bucket, float* __restrict__ counts) {
    int e = blockIdx.x * 8 + (threadIdx.x >> 5);
    int lane = threadIdx.x & 31;
    if (e >= E_EDGES) return;
    int s = src[e], t = tgt[e], ty = et[e];
    float*       dst = bucket + ((size_t)ty * N_NODES + t) * DIM;
    const float* sf  = x + (size_t)s * DIM;
    for (int d = lane; d < DIM; d += 32) atomicAdd(&dst[d], sf[d]);
    if (lane == 0) atomicAdd(&counts[ty * N_NODES + t], 1.0f);
}

// ---------------- agg = sum_t bucket[t] @ W[t]^T + counts*b ----------------
__global__ void __launch_bounds__(256) k_agg_gemm(const float* __restrict__ bucket,
                                                  const float* __restrict__ counts,
                                                  const float* __restrict__ msgW,
                                                  const float* __restrict__ msgb,
                                                  float* __restrict__ agg) {
    int wave = threadIdx.x >> 5, lane = threadIdx.x & 31;
    int tile = blockIdx.x * 8 + wave;
    int r0 = (tile / (DIM / 16)) * 16;
    int c0 = (tile % (DIM / 16)) * 16;
    int m = lane & 15, kh = lane >> 4;
    v8f acc = {};
    for (int t = 0; t < NET; ++t) {
        const float* A = bucket + (size_t)t * N_NODES * DIM + (size_t)(r0 + m) * DIM;
        const float* B = msgW + (size_t)t * DIM * DIM + (size_t)(c0 + m) * DIM;
        for (int k = 0; k < DIM; k += 4) {
            int ka = k + 2 * kh;
            acc = wmma4(A[ka], A[ka + 1], B[ka], B[ka + 1], acc);
        }
    }
    for (int j = 0; j < 8; ++j) {
        int row = r0 + j + 8 * kh;
        int col = c0 + m;
        float v = acc[j];
        for (int t = 0; t < NET; ++t)
            v += counts[t * N_NODES + row] * msgb[t * DIM + col];
        agg[(size_t)row * DIM + col] = v;
    }
}

// ---------------- h = x + relu([x,agg] @ updW^T + b) ----------------
__global__ void __launch_bounds__(256) k_upd_gemm(const float* __restrict__ x,
                                                  const float* __restrict__ agg,
                                                  const float* __restrict__ updW,
                                                  const float* __restrict__ updb,
                                                  float* __restrict__ h) {
    int wave = threadIdx.x >> 5, lane = threadIdx.x & 31;
    int tile = blockIdx.x * 8 + wave;
    int r0 = (tile / (DIM / 16)) * 16;
    int c0 = (tile % (DIM / 16)) * 16;
    int m = lane & 15, kh = lane >> 4;
    const float* Wr = updW + (size_t)(c0 + m) * (2 * DIM);
    const float* A1 = x   + (size_t)(r0 + m) * DIM;
    const float* A2 = agg + (size_t)(r0 + m) * DIM;
    const float* W2 = Wr + DIM;
    v8f acc = {};
    for (int k = 0; k < DIM; k += 4) {
        int ka = k + 2 * kh;
        acc = wmma4(A1[ka], A1[ka + 1], Wr[ka], Wr[ka + 1], acc);
    }
    for (int k = 0; k < DIM; k += 4) {
        int ka = k + 2 * kh;
        acc = wmma4(A2[ka], A2[ka + 1], W2[ka], W2[ka + 1], acc);
    }
    for (int j = 0; j < 8; ++j) {
        int row = r0 + j + 8 * kh;
        int col = c0 + m;
        float u = fmaxf(acc[j] + updb[col], 0.0f);
        h[(size_t)row * DIM + col] = x[(size_t)row * DIM + col] + u;
    }
}

// ---------------- layernorm per row ----------------
__global__ void k_ln(const float* __restrict__ h, const float* __restrict__ g,
                     const float* __restrict__ b, float* __restrict__ out) {
    __shared__ float red[256];
    int row = blockIdx.x;
    const float* hr = h + (size_t)row * DIM;
    float s = 0.f;
    for (int c = threadIdx.x; c < DIM; c += 256) s += hr[c];
    red[threadIdx.x] = s; __syncthreads();
    for (int o = 128; o > 0; o >>= 1) { if (threadIdx.x < o) red[threadIdx.x] += red[threadIdx.x + o]; __syncthreads(); }
    float mu = red[0] / DIM; __syncthreads();
    float v = 0.f;
    for (int c = threadIdx.x; c < DIM; c += 256) { float d = hr[c] - mu; v += d * d; }
    red[threadIdx.x] = v; __syncthreads();
    for (int o = 128; o > 0; o >>= 1) { if (threadIdx.x < o) red[threadIdx.x] += red[threadIdx.x + o]; __syncthreads(); }
    float inv = rsqrtf(red[0] / DIM + 1e-5f);
    for (int c = threadIdx.x; c < DIM; c += 256)
        out[(size_t)row * DIM + c] = g[c] * (hr[c] - mu) * inv + b[c];
}

// ---------------- qkv = x @ inW^T + b ----------------
__global__ void __launch_bounds__(256) k_qkv(const float* __restrict__ x,
                                             const float* __restrict__ W,
                                             const float* __restrict__ bias,
                                             float* __restrict__ qkv) {
    int wave = threadIdx.x >> 5, lane = threadIdx.x & 31;
    int tile = blockIdx.x * 8 + wave;
    int r0 = (tile / (TD / 16)) * 16;
    int c0 = (tile % (TD / 16)) * 16;
    int m = lane & 15, kh = lane >> 4;
    const float* A = x + (size_t)(r0 + m) * DIM;
    const float* B = W + (size_t)(c0 + m) * DIM;
    v8f acc = {};
    for (int k = 0; k < DIM; k += 4) {
        int ka = k + 2 * kh;
        acc = wmma4(A[ka], A[ka + 1], B[ka], B[ka + 1], acc);
    }
    for (int j = 0; j < 8; ++j) {
        int row = r0 + j + 8 * kh;
        int col = c0 + m;
        qkv[(size_t)row * TD + col] = acc[j] + bias[col];
    }
}

// ---------------- flash attention, 16 query rows per block ----------------
__global__ void __launch_bounds__(256) k_attn(const float* __restrict__ qkv,
                                              float* __restrict__ attnw,
                                              float* __restrict__ ctxbuf) {
    __shared__ float sS[16 * CH];
    __shared__ float sRed[16][17];
    __shared__ float sM[16], sZ[16], sInvZ[16];
    int q0 = blockIdx.x * 16;
    int wave = threadIdx.x >> 5, lane = threadIdx.x & 31;
    int m = lane & 15, kh = lane >> 4;
    int rr = threadIdx.x >> 4;     // 16 rows x 16 threads
    int cc = threadIdx.x & 15;
    const float scale = 0.10206207261596577f;  // 1/sqrt(96)

    for (int h = 0; h < NH; ++h) {
        if (threadIdx.x < 16) { sM[threadIdx.x] = -1e30f; sZ[threadIdx.x] = 0.0f; }
        __syncthreads();
        // ---- pass 1: online row max / sum ----
        for (int ch = 0; ch < N_NODES; ch += CH) {
            for (int tt = wave; tt < CH / 16; tt += 8) {
                int kc0 = ch + tt * 16;
                const float* A = qkv + (size_t)(q0 + m) * TD + h * DH;
                const float* B = qkv + (size_t)(kc0 + m) * TD + DIM + h * DH;
                v8f acc = {};
                for (int k = 0; k < DH; k += 4) {
                    int ka = k + 2 * kh;
                    acc = wmma4(A[ka], A[ka + 1], B[ka], B[ka + 1], acc);
                }
                for (int j = 0; j < 8; ++j)
                    sS[(j + 8 * kh) * CH + tt * 16 + m] = acc[j] * scale;
            }
            __syncthreads();
            float lmax = -1e30f;
            for (int c = cc; c < CH; c += 16) lmax = fmaxf(lmax, sS[rr * CH + c]);
            sRed[rr][cc] = lmax; __syncthreads();
            if (cc == 0) {
                float cm = sRed[rr][0];
                for (int i = 1; i < 16; ++i) cm = fmaxf(cm, sRed[rr][i]);
                float nm = fmaxf(sM[rr], cm);
                sZ[rr] *= __expf(sM[rr] - nm);
                sM[rr] = nm;
            }
            __syncthreads();
            float mrow = sM[rr];
            float lsum = 0.f;
            for (int c = cc; c < CH; c += 16) lsum += __expf(sS[rr * CH + c] - mrow);
            sRed[rr][cc] = lsum; __syncthreads();
            if (cc == 0) { float t = 0.f; for (int i = 0; i < 16; ++i) t += sRed[rr][i]; sZ[rr] += t; }
            __syncthreads();
        }
        if (threadIdx.x < 16) sInvZ[threadIdx.x] = 1.0f / sZ[threadIdx.x];
        __syncthreads();
        // ---- pass 2: normalize, emit attn mean, accumulate P@V ----
        v8f cacc = {};
        for (int ch = 0; ch < N_NODES; ch += CH) {
            for (int tt = wave; tt < CH / 16; tt += 8) {
                int kc0 = ch + tt * 16;
                const float* A = qkv + (size_t)(q0 + m) * TD + h * DH;
                const float* B = qkv + (size_t)(kc0 + m) * TD + DIM + h * DH;
                v8f acc = {};
                for (int k = 0; k < DH; k += 4) {
                    int ka = k + 2 * kh;
                    acc = wmma4(A[ka], A[ka + 1], B[ka], B[ka + 1], acc);
                }
                for (int j = 0; j < 8; ++j)
                    sS[(j + 8 * kh) * CH + tt * 16 + m] = acc[j] * scale;
            }
            __syncthreads();
            for (int idx = threadIdx.x; idx < 16 * CH; idx += 256) {
                int r = idx / CH;
                sS[idx] = __expf(sS[idx] - sM[r]) * sInvZ[r];
            }
            __syncthreads();
            for (int idx = threadIdx.x; idx < 16 * CH; idx += 256) {
                int r = idx / CH, c = idx % CH;
                size_t off = (size_t)(q0 + r) * N_NODES + ch + c;
                float val = sS[idx] * 0.125f;
                if (h == 0) attnw[off] = val; else attnw[off] += val;
            }
            if (wave < 6) {  // 6 x 16 columns cover DH=96; wave-uniform branch (EXEC stays full)
                const float* Bv = qkv + (size_t)ch * TD + 2 * DIM + h * DH + wave * 16 + m;
                const float* Ap = &sS[m * CH];
                for (int k = 0; k < CH; k += 4) {
                    int ka = k + 2 * kh;
                    cacc = wmma4(Ap[ka], Ap[ka + 1], Bv[(size_t)ka * TD], Bv[(size_t)(ka + 1) * TD], cacc);
                }
            }
            __syncthreads();
        }
        if (wave < 6) {
            for (int j = 0; j < 8; ++j) {
                int row = q0 + j + 8 * kh;
                int col = h * DH + wave * 16 + m;
                ctxbuf[(size_t)row * DIM + col] = cacc[j];
            }
        }
        __syncthreads();
    }
}

// ---------------- graph_repr = ctx @ outW^T + b ----------------
__global__ void __launch_bounds__(256) k_outproj(const float* __restrict__ ctx,
                                                 const float* __restrict__ W,
                                                 const float* __restrict__ bias,
                                                 float* __restrict__ out) {
    int wave = threadIdx.x >> 5, lane = threadIdx.x & 31;
    int tile = blockIdx.x * 8 + wave;
    int r0 = (tile / (DIM / 16)) * 16;
    int c0 = (tile % (DIM / 16)) * 16;
    int m = lane & 15, kh = lane >> 4;
    const float* A = ctx + (size_t)(r0 + m) * DIM;
    const float* B = W + (size_t)(c0 + m) * DIM;
    v8f acc = {};
    for (int k = 0; k < DIM; k += 4) {
        int ka = k + 2 * kh;
        acc = wmma4(A[ka], A[ka + 1], B[ka], B[ka + 1], acc);
    }
    for (int j = 0; j < 8; ++j) {
        int row = r0 + j + 8 * kh;
        int col = c0 + m;
        out[(size_t)row * DIM + col] = acc[j] + bias[col];
    }
}

// ---------------- pooled mean -> 3-layer MLP -> sigmoid ----------------
__global__ void k_mlp(const float* __restrict__ gr,
                      const float* __restrict__ W1, const float* __restrict__ b1,
                      const float* __restrict__ W2, const float* __restrict__ b2,
                      const float* __restrict__ W3, const float* __restrict__ b3,
                      float* __restrict__ score) {
    __shared__ float pooled[DIM];
    __shared__ float h1[512];
    __shared__ float h2[256];
    __shared__ float red[256];
    for (int c = threadIdx.x; c < DIM; c += 256) {
        float s = 0.f;
        for (int r = 0; r < N_NODES; ++r) s += gr[(size_t)r * DIM + c];
        pooled[c] = s / (float)N_NODES;
    }
    __syncthreads();
    for (int o = threadIdx.x; o < 512; o += 256) {
        float s = b1[o];
        const float* w = W1 + (size_t)o * DIM;
        for (int c = 0; c < DIM; ++c) s += w[c] * pooled[c];
        h1[o] = fmaxf(s, 0.f);
    }
    __syncthreads();
    {
        int o = threadIdx.x;
        float s = b2[o];
        const float* w = W2 + (size_t)o * 512;
        for (int c = 0; c < 512; ++c) s += w[c] * h1[c];
        h2[o] = fmaxf(s, 0.f);
    }
    __syncthreads();
    red[threadIdx.x] = W3[threadIdx.x] * h2[threadIdx.x];
    __syncthreads();
    for (int o = 128; o > 0; o >>= 1) { if (threadIdx.x < o) red[threadIdx.x] += red[threadIdx.x + o]; __syncthreads(); }
    if (threadIdx.x == 0) {
        float z = red[0] + b3[0];
        score[0] = 1.0f / (1.0f + __expf(-z));
    }
}

extern "C" void kernel_launch(void* const* d_in, const int* in_sizes, int n_in,
                              void* d_out, int out_size, void* d_ws, size_t ws_size,
                              hipStream_t stream) {
    const float* nf   = (const float*)d_in[0];
    const int*   ei   = (const int*)d_in[1];
    const int*   et   = (const int*)d_in[2];
    const int*   nt   = (const int*)d_in[3];
    const float* st   = (const float*)d_in[4];
    const float* ca   = (const float*)d_in[5];
    const float* ar   = (const float*)d_in[6];
    const float* msgW = (const float*)d_in[7];
    const float* msgb = (const float*)d_in[8];
    const float* updW = (const float*)d_in[9];
    const float* updb = (const float*)d_in[10];
    const float* lng  = (const float*)d_in[11];
    const float* lnb  = (const float*)d_in[12];
    const float* ipW  = (const float*)d_in[13];
    const float* ipb  = (const float*)d_in[14];
    const float* opW  = (const float*)d_in[15];
    const float* opb  = (const float*)d_in[16];
    const float* W1   = (const float*)d_in[17];
    const float* b1   = (const float*)d_in[18];
    const float* W2   = (const float*)d_in[19];
    const float* b2   = (const float*)d_in[20];
    const float* W3   = (const float*)d_in[21];
    const float* b3   = (const float*)d_in[22];

    const size_t ND = (size_t)N_NODES * DIM;
    float* ws     = (float*)d_ws;
    float* xa     = ws;
    float* xb     = xa + ND;
    float* agg    = xb + ND;
    float* qkv    = agg + ND;
    float* ctx    = qkv + (size_t)N_NODES * TD;
    float* bucket = ctx + ND;
    float* counts = bucket + (size_t)NET * ND;

    float* out        = (float*)d_out;
    float* out_x      = out;
    float* out_gr     = out + ND;
    float* out_score  = out + 2 * ND;
    float* out_attnw  = out + 2 * ND + 1;

    k_embed<<<N_NODES, 256, 0, stream>>>(nf, nt, st, ca, ar, xa);

    const float* xcur = xa;
    for (int l = 0; l < NL; ++l) {
        hipMemsetAsync(bucket, 0, ((size_t)NET * ND + NET * N_NODES) * sizeof(float), stream);
        k_scatter<<<E_EDGES / 8, 256, 0, stream>>>(xcur, ei, ei + E_EDGES, et, bucket, counts);
        k_agg_gemm<<<(N_NODES / 16) * (DIM / 16) / 8, 256, 0, stream>>>(
            bucket, counts, msgW + (size_t)l * NET * DIM * DIM, msgb + (size_t)l * NET * DIM, agg);
        float* hbuf = bucket;  // bucket space is free until next layer's memset
        k_upd_gemm<<<(N_NODES / 16) * (DIM / 16) / 8, 256, 0, stream>>>(
            xcur, agg, updW + (size_t)l * DIM * 2 * DIM, updb + (size_t)l * DIM, hbuf);
        float* dst = (l == 0) ? xb : (l == 1) ? xa : out_x;  // final layer lands in d_out
        k_ln<<<N_NODES, 256, 0, stream>>>(hbuf, lng + (size_t)l * DIM, lnb + (size_t)l * DIM, dst);
        xcur = dst;
    }

    k_qkv<<<(N_NODES / 16) * (TD / 16) / 8, 256, 0, stream>>>(xcur, ipW, ipb, qkv);
    k_attn<<<N_NODES / 16, 256, 0, stream>>>(qkv, out_attnw, ctx);
    k_outproj<<<(N_NODES / 16) * (DIM / 16) / 8, 256, 0, stream>>>(ctx, opW, opb, out_gr);
    k_mlp<<<1, 256, 0, stream>>>(out_gr, W1, b1, W2, b2, W3, b3, out_score);
}